// SpectralConv2d_57492432224352
// MI455X (gfx1250) — compile-verified
//
#include <hip/hip_runtime.h>
#include <hip/hip_bf16.h>
#include <math.h>
#include <stdint.h>

// ---------------------------------------------------------------------------
// SpectralConv2d on MI455X (gfx1250). All matmuls via V_WMMA_F32_16X16X4_F32;
// x is streamed HBM->LDS with gfx1250 async-LDS copies (ASYNCcnt).
//
//   K1 (fused): x chunks async-staged into LDS (double buffered);
//               Y = F(16x256) * x(256x256) per image (16 waves);
//               Z = Y * F^T -> 16x16 complex modes (wave 0).
//   K3:         S = Z * W (channel mix per mode, complex).
//   K45 (fused): T = G(256x16)*S in LDS (per-wave lane transpose),
//               out = Re(T * G2^T) -> 256x256, written once to HBM.
//
// Roofline: ~5 GFLOP vs 268 MB mandatory HBM traffic -> memory bound,
// ~11.5 us floor @ 23.3 TB/s. Intermediates (~3 MB) are L2-resident.
// ---------------------------------------------------------------------------

typedef __attribute__((ext_vector_type(2))) float v2f;
typedef __attribute__((ext_vector_type(8))) float v8f;

#define PI2_256 0.02454369260617026f   // 2*pi/256
#define INV_N2  (1.0f / 65536.0f)      // ifft2 normalization 1/(256*256)

// ---- workspace layout (float offsets) ----
#define WS_FWD_R   0        // [16][256] cos(2pi*m*h/256)
#define WS_FWD_I   4096     // [16][256] -sin
#define WS_INVA_R  8192     // [256][16] cos * 1/65536
#define WS_INVA_I  12288    // [256][16] +sin * 1/65536
#define WS_INVB_R  16384    // [16][256] cos
#define WS_INVB_IN 20480    // [16][256] -sin
#define WS_ZR      24576    // [B=16][Cin=32][16][16]
#define WS_ZI      155648
#define WS_SR      286720   // [B][Cout=32][16][16]
#define WS_SI      417792
// total = 548864 floats ~ 2.1 MB

// ---- WMMA helper: D = A(16x4) * B(4x16) + C, f32 ----
__device__ __forceinline__ v8f wmma4(v2f a, v2f b, v8f c) {
  return __builtin_amdgcn_wmma_f32_16x16x4_f32(
      /*neg_a=*/false, a, /*neg_b=*/false, b,
      /*c_mod=*/(short)0, c, /*reuse_a=*/false, /*reuse_b=*/false);
}

// A fragment: row-major A[16][lda], K-slice starting at k0
__device__ __forceinline__ v2f ldA(const float* A, int lda, int lm, int half, int k0) {
  int k = k0 + 2 * half;
  v2f r;
  r.x = A[lm * lda + k];
  r.y = A[lm * lda + k + 1];
  return r;
}

// ---- gfx1250 async global->LDS copy (ASYNCcnt-tracked), 16B per lane ----
__device__ __forceinline__ void async_ld_b128(uint32_t lds_off, const void* gaddr) {
  asm volatile("global_load_async_to_lds_b128 %0, %1, off"
               :: "v"(lds_off), "v"(gaddr) : "memory");
}
__device__ __forceinline__ void wait_async0() {
  asm volatile("s_wait_asynccnt 0x0" ::: "memory");
}

// ---------------------------------------------------------------------------
// K0: twiddle tables (computed on-device each call; graph-capture safe)
// ---------------------------------------------------------------------------
__global__ void twiddle_init_kernel(float* __restrict__ ws) {
  int t = blockIdx.x * blockDim.x + threadIdx.x;   // 0..4095
  if (t >= 4096) return;
  int a = t >> 8;    // mode index 0..15
  int b = t & 255;   // position  0..255
  int ph = (a * b) & 255;                          // exact angle reduction
  float th = (float)ph * PI2_256;
  float s = sinf(th), c = cosf(th);
  ws[WS_FWD_R   + a * 256 + b] = c;
  ws[WS_FWD_I   + a * 256 + b] = -s;
  ws[WS_INVA_R  + b * 16  + a] = c * INV_N2;
  ws[WS_INVA_I  + b * 16  + a] = s * INV_N2;
  ws[WS_INVB_R  + a * 256 + b] = c;
  ws[WS_INVB_IN + a * 256 + b] = -s;
}

// ---------------------------------------------------------------------------
// K1 (fused forward): per image (b,i).
//   x staged HBM->LDS in 16-row (16KB) chunks, double buffered, async.
//   stage 1: 16 waves each produce one 16-column tile of Y = F*x  (LDS)
//   stage 2: wave 0 reduces Z = Y * F^T  (16x16 complex modes)
// x is read from HBM exactly once, as contiguous b128-per-lane async streams.
// ---------------------------------------------------------------------------
__global__ __launch_bounds__(512) void fwd_dft_kernel(const float* __restrict__ x,
                                                      float* __restrict__ ws) {
  __shared__ float Xbuf[2 * 4096];                 // 2 x (16 rows x 256 cols)
  __shared__ float Ylds[2 * 4096];                 // Yr[16][256], Yi[16][256]
  const float* FWD_R = ws + WS_FWD_R;
  const float* FWD_I = ws + WS_FWD_I;
  float* Zr = ws + WS_ZR;
  float* Zi = ws + WS_ZI;

  int img  = blockIdx.x;                           // 0..511 (= b*32 + i)
  const char* Xc = (const char*)(x + (size_t)img * 65536);
  int tid  = threadIdx.x;
  int lane = tid & 31;
  int wv   = tid >> 5;
  int lm   = lane & 15;
  int half = lane >> 4;

  uint32_t xb = (uint32_t)(uintptr_t)(void*)Xbuf;  // LDS byte offset of Xbuf

  // prologue: async-stage chunk 0 (each thread moves 2 x 16B)
  {
    uint32_t l = xb + tid * 16;
    async_ld_b128(l,        Xc + tid * 16);
    async_ld_b128(l + 8192, Xc + tid * 16 + 8192);
  }
  wait_async0();
  __syncthreads();

  // ---- stage 1: Y tile (16 modes x 16 cols) per wave, chunked over h ----
  v8f accR = {}, accI = {};
  int col0 = wv * 16;
  for (int c = 0; c < 16; ++c) {
    if (c < 15) {                                  // prefetch next chunk
      uint32_t l = xb + ((c + 1) & 1) * 16384 + tid * 16;
      const char* g = Xc + (c + 1) * 16384 + tid * 16;
      async_ld_b128(l, g);
      async_ld_b128(l + 8192, g + 8192);
    }
    const float* Xb = Xbuf + (c & 1) * 4096;       // current chunk in LDS
    for (int kk = 0; kk < 16; kk += 4) {
      int kg = c * 16 + kk;                        // global K index (row h)
      v2f aR = ldA(FWD_R, 256, lm, half, kg);
      v2f aI = ldA(FWD_I, 256, lm, half, kg);
      int kr = kk + 2 * half;
      v2f bX;
      bX.x = Xb[kr * 256 + col0 + lm];
      bX.y = Xb[(kr + 1) * 256 + col0 + lm];
      accR = wmma4(aR, bX, accR);
      accI = wmma4(aI, bX, accI);
    }
    if (c < 15) { wait_async0(); __syncthreads(); }
  }
  for (int v = 0; v < 8; ++v) {
    int row = v + 8 * half;
    Ylds[row * 256 + col0 + lm]        = accR[v];
    Ylds[4096 + row * 256 + col0 + lm] = accI[v];
  }
  __syncthreads();

  // ---- stage 2: Z = Y * F^T (wave 0 only; EXEC all-1 within the wave) ----
  if (wv == 0) {
    const float* Yr = Ylds;
    const float* Yi = Ylds + 4096;
    v8f aRR = {}, aII = {}, aRI = {}, aIR = {};
    for (int k0 = 0; k0 < 256; k0 += 4) {
      v2f yR = ldA(Yr, 256, lm, half, k0);
      v2f yI = ldA(Yi, 256, lm, half, k0);
      // B[k=w][n] = table[n][w]  (symmetric index -> same access as A-frag)
      v2f fR = ldA(FWD_R, 256, lm, half, k0);
      v2f fI = ldA(FWD_I, 256, lm, half, k0);
      aRR = wmma4(yR, fR, aRR);
      aII = wmma4(yI, fI, aII);
      aRI = wmma4(yR, fI, aRI);
      aIR = wmma4(yI, fR, aIR);
    }
    size_t base = (size_t)img * 256;
    for (int v = 0; v < 8; ++v) {
      int m = v + 8 * half;
      Zr[base + m * 16 + lm] = aRR[v] - aII[v];
      Zi[base + m * 16 + lm] = aRI[v] + aIR[v];
    }
  }
}

// ---------------------------------------------------------------------------
// K3: channel mix, per mode: S[b,o] = sum_i Z[b,i] * W[i,o]  (complex)
// One wave per (mode, o-tile): 512 waves. All data L2-resident.
// ---------------------------------------------------------------------------
__global__ __launch_bounds__(256) void mix_kernel(const float* __restrict__ wr,
                                                  const float* __restrict__ wi,
                                                  float* __restrict__ ws) {
  const float* Zr = ws + WS_ZR;
  const float* Zi = ws + WS_ZI;
  float* Sr = ws + WS_SR;
  float* Si = ws + WS_SI;

  int W    = blockIdx.x * 8 + (threadIdx.x >> 5); // 0..511
  int lane = threadIdx.x & 31;
  int lm   = lane & 15;
  int half = lane >> 4;
  int mode = W >> 1;                              // 0..255 (= m*16 + n)
  int ot   = W & 1;                               // o-tile
  int o    = ot * 16 + lm;

  v8f aRR = {}, aII = {}, aRI = {}, aIR = {};
  for (int k0 = 0; k0 < 32; k0 += 4) {
    int k = k0 + 2 * half;
    v2f xr, xi, br, bi;
    xr.x = Zr[(lm * 32 + k) * 256 + mode];
    xr.y = Zr[(lm * 32 + k + 1) * 256 + mode];
    xi.x = Zi[(lm * 32 + k) * 256 + mode];
    xi.y = Zi[(lm * 32 + k + 1) * 256 + mode];
    br.x = wr[(k * 32 + o) * 256 + mode];
    br.y = wr[((k + 1) * 32 + o) * 256 + mode];
    bi.x = wi[(k * 32 + o) * 256 + mode];
    bi.y = wi[((k + 1) * 32 + o) * 256 + mode];
    aRR = wmma4(xr, br, aRR);
    aII = wmma4(xi, bi, aII);
    aRI = wmma4(xr, bi, aRI);
    aIR = wmma4(xi, br, aIR);
  }
  for (int v = 0; v < 8; ++v) {
    int b = v + 8 * half;
    Sr[(b * 32 + o) * 256 + mode] = aRR[v] - aII[v];
    Si[(b * 32 + o) * 256 + mode] = aRI[v] + aIR[v];
  }
}

// ---------------------------------------------------------------------------
// K45 (fused inverse): per image (b,o), 16 waves.
//   stage A: wave w computes h-tile w of T(256x16) = G*S (complex),
//            stages it in LDS (this is just the D-layout -> A-layout lane
//            transpose, produced and consumed by the SAME wave: no barrier).
//   stage B: wave w loops the 16 w-tiles of its h-strip:
//            out = Tr*cos + Ti*(-sin), single accumulator, 2 WMMA per k-step.
// out is written to HBM exactly once.
// ---------------------------------------------------------------------------
__global__ __launch_bounds__(512) void inv_dft_kernel(const float* __restrict__ ws,
                                                      float* __restrict__ out) {
  __shared__ float Tl[2 * 4096];                   // Tr[256][16], Ti[256][16]
  const float* GA_R = ws + WS_INVA_R;              // [256][16]
  const float* GA_I = ws + WS_INVA_I;
  const float* BR   = ws + WS_INVB_R;              // [16][256] cos
  const float* BIN  = ws + WS_INVB_IN;             // [16][256] -sin
  const float* Srb  = ws + WS_SR + (size_t)blockIdx.x * 256;
  const float* Sib  = ws + WS_SI + (size_t)blockIdx.x * 256;

  int lane = threadIdx.x & 31;
  int ht   = threadIdx.x >> 5;                     // h-tile = wave id, 0..15
  int lm   = lane & 15;
  int half = lane >> 4;

  // ---- stage A: T h-tile = G * S ----
  {
    v8f aRR = {}, aII = {}, aRI = {}, aIR = {};
    for (int k0 = 0; k0 < 16; k0 += 4) {
      int k = k0 + 2 * half;
      int h = ht * 16 + lm;
      v2f gr, gi, sr, si;
      gr.x = GA_R[h * 16 + k];  gr.y = GA_R[h * 16 + k + 1];
      gi.x = GA_I[h * 16 + k];  gi.y = GA_I[h * 16 + k + 1];
      sr.x = Srb[k * 16 + lm];  sr.y = Srb[(k + 1) * 16 + lm];
      si.x = Sib[k * 16 + lm];  si.y = Sib[(k + 1) * 16 + lm];
      aRR = wmma4(gr, sr, aRR);
      aII = wmma4(gi, si, aII);
      aRI = wmma4(gr, si, aRI);
      aIR = wmma4(gi, sr, aIR);
    }
    for (int v = 0; v < 8; ++v) {
      int h = ht * 16 + v + 8 * half;
      Tl[h * 16 + lm]        = aRR[v] - aII[v];
      Tl[4096 + h * 16 + lm] = aRI[v] + aIR[v];
    }
  }
  // same wave produced the rows it consumes below -> DScnt ordering suffices

  // ---- stage B: hoist T fragments once, sweep the 16 w-tiles ----
  v2f tr[4], ti[4];
  for (int ks = 0; ks < 4; ++ks) {
    int k = ks * 4 + 2 * half;
    int h = ht * 16 + lm;
    tr[ks].x = Tl[h * 16 + k];        tr[ks].y = Tl[h * 16 + k + 1];
    ti[ks].x = Tl[4096 + h * 16 + k]; ti[ks].y = Tl[4096 + h * 16 + k + 1];
  }
  size_t base = (size_t)blockIdx.x * 65536;
  for (int wt = 0; wt < 16; ++wt) {
    int w = wt * 16 + lm;
    v8f acc = {};
    for (int ks = 0; ks < 4; ++ks) {
      int k = ks * 4 + 2 * half;
      v2f br, bi;
      br.x = BR[k * 256 + w];   br.y = BR[(k + 1) * 256 + w];
      bi.x = BIN[k * 256 + w];  bi.y = BIN[(k + 1) * 256 + w];
      acc = wmma4(tr[ks], br, acc);
      acc = wmma4(ti[ks], bi, acc);
    }
    for (int v = 0; v < 8; ++v) {
      int h = ht * 16 + v + 8 * half;
      out[base + h * 256 + wt * 16 + lm] = acc[v];
    }
  }
}

// ---------------------------------------------------------------------------
extern "C" void kernel_launch(void* const* d_in, const int* in_sizes, int n_in,
                              void* d_out, int out_size, void* d_ws, size_t ws_size,
                              hipStream_t stream) {
  (void)in_sizes; (void)n_in; (void)out_size; (void)ws_size;
  const float* x  = (const float*)d_in[0];   // [16][32][256][256]
  const float* wr = (const float*)d_in[1];   // [32][32][16][16]
  const float* wi = (const float*)d_in[2];   // [32][32][16][16]
  float* ws  = (float*)d_ws;
  float* out = (float*)d_out;                // [16][32][256][256]

  twiddle_init_kernel<<<16, 256, 0, stream>>>(ws);
  fwd_dft_kernel<<<512, 512, 0, stream>>>(x, ws);   // 512 images, 16 waves each
  mix_kernel<<<64, 256, 0, stream>>>(wr, wi, ws);   // 512 waves
  inv_dft_kernel<<<512, 512, 0, stream>>>(ws, out); // 512 images, 16 waves each
}